// FISLayer_3719441679094
// MI455X (gfx1250) — compile-verified
//
#include <hip/hip_runtime.h>
#include <stdint.h>
#include <math.h>

// Problem constants (match reference)
#define T_TREES 8
#define T_NODES 5
#define T_C     16
#define T_B     8
#define T_H     224
#define T_W     224
#define NCHUNK  7            // 224 / 32 lanes
#define NEGV    (-1e30f)

struct ForestParams { int par[T_TREES][T_NODES]; };

// ---------------------------------------------------------------------------
// Cross-lane helpers in the VALU pipe (DPP16 + readlane), wave32
// ---------------------------------------------------------------------------
template<int CTRL>
__device__ __forceinline__ float dpp_f(float x) {
  int xi = __float_as_int(x);
  // old = src: lanes whose DPP source is out-of-row keep their own value
  int r = __builtin_amdgcn_update_dpp(xi, xi, CTRL, 0xF, 0xF, false);
  return __int_as_float(r);
}
__device__ __forceinline__ float readlane_f(float x, int l) {
  return __int_as_float(__builtin_amdgcn_readlane(__float_as_int(x), l));
}

// Inclusive 32-lane cummax: 4x DPP row_shr within 16-lane rows + cross-half fix
__device__ __forceinline__ float wave_incl_cummax(float v, int lane) {
  v = fmaxf(v, dpp_f<0x111>(v));   // row_shr:1
  v = fmaxf(v, dpp_f<0x112>(v));   // row_shr:2
  v = fmaxf(v, dpp_f<0x114>(v));   // row_shr:4
  v = fmaxf(v, dpp_f<0x118>(v));   // row_shr:8
  float l15 = readlane_f(v, 15);   // prefix of lower half
  float hi  = fmaxf(v, l15);
  return (lane & 16) ? hi : v;
}

// ---------------------------------------------------------------------------
// CDNA5 async copy (global -> LDS) with double buffering
// ---------------------------------------------------------------------------
__device__ __forceinline__ void async_row_load(const float* __restrict__ g,
                                               float* lds, int lane) {
#if defined(__gfx1250__)
  uint32_t lbase = (uint32_t)(uintptr_t)lds;          // low 32 bits = LDS offset
#pragma unroll
  for (int k = 0; k < NCHUNK; ++k) {
    uint32_t la = lbase + (uint32_t)((k * 32 + lane) * 4);
    uint64_t ga = (uint64_t)(uintptr_t)(g + k * 32 + lane);
    asm volatile("global_load_async_to_lds_b32 %0, %1, off"
                 :: "v"(la), "v"(ga) : "memory");
  }
#else
#pragma unroll
  for (int k = 0; k < NCHUNK; ++k) lds[k * 32 + lane] = g[k * 32 + lane];
#endif
}

__device__ __forceinline__ void wait_async_le7() {
#if defined(__gfx1250__)
  asm volatile("s_wait_asynccnt 0x7" ::: "memory");
#endif
}
__device__ __forceinline__ void wait_async_0() {
#if defined(__gfx1250__)
  asm volatile("s_wait_asynccnt 0x0" ::: "memory");
#endif
}

// ---------------------------------------------------------------------------
// Kernel: one block per (b,c) plane, one wave (32 lanes) per tree.
// Row-streaming formulation of the nested strict 2D prefix-max recursion:
//   colmax_j[w] = inclusive 2D cummax of rows processed so far;
//   epm_row[w]  = colmax_j[w-1] (pre-update), NEG on boundaries.
// ---------------------------------------------------------------------------
__global__ __launch_bounds__(256, 1)
void fis_forest_kernel(const float* __restrict__ x,
                       const float* __restrict__ alphas,
                       float* __restrict__ out,
                       ForestParams fp) {
  __shared__ float s_x[T_TREES][2][T_W];   // per-wave double buffer (14.3 KB)

  const int lane  = threadIdx.x & 31;
  const int t     = threadIdx.x >> 5;      // tree id = wave id (0..7)
  const int plane = blockIdx.x;            // 0..127
  const int b     = plane >> 4;
  const int c     = plane & 15;

  const float* xp = x + ((size_t)b * T_C + c) * (size_t)(T_H * T_W);

  // alpha[t][j][c] scalars (wave-uniform)
  float a[T_NODES];
#pragma unroll
  for (int j = 0; j < T_NODES; ++j)
    a[j] = alphas[(t * T_NODES + j) * T_C + c];

  int par[T_NODES];
#pragma unroll
  for (int j = 0; j < T_NODES; ++j) par[j] = fp.par[t][j];

  // colmax state per node 1..4
  float cm[T_NODES][NCHUNK];
#pragma unroll
  for (int j = 0; j < T_NODES; ++j)
#pragma unroll
    for (int k = 0; k < NCHUNK; ++k) cm[j][k] = NEGV;

  float outm = NEGV;

  // prefetch row 0 into buffer 0
  async_row_load(xp, &s_x[t][0][0], lane);

  for (int h = 0; h < T_H; ++h) {
    // prefetch next row into the other buffer (dummy re-read of row 0 at end)
    const int nh = (h + 1 == T_H) ? 0 : (h + 1);
    async_row_load(xp + (size_t)nh * T_W, &s_x[t][(h + 1) & 1][0], lane);
    wait_async_le7();                       // row h resident in LDS

    // xv = log1p(relu(x)) for this row (7 chunks of 32 lanes)
    float xv[NCHUNK];
#pragma unroll
    for (int k = 0; k < NCHUNK; ++k) {
      float v = s_x[t][h & 1][k * 32 + lane];
      xv[k] = log1pf(fmaxf(v, 0.0f));
    }

    // per-row child-accumulator for every node (sum of children epm rows)
    float acc[T_NODES][NCHUNK];
#pragma unroll
    for (int j = 0; j < T_NODES; ++j)
#pragma unroll
      for (int k = 0; k < NCHUNK; ++k) acc[j][k] = 0.0f;

    // nodes 4..1 (children before parents since par[j] < j)
#pragma unroll
    for (int jj = 0; jj < T_NODES - 1; ++jj) {
      const int j = T_NODES - 1 - jj;

      // v_j row
      float v[NCHUNK];
#pragma unroll
      for (int k = 0; k < NCHUNK; ++k) v[k] = (xv[k] + a[j]) + acc[j][k];

      // epm row = old colmax shifted right by one column, NEG at col 0.
      // DPP row_shr:1 handles all lanes except 0 and 16; fix those with
      // readlane broadcasts (lane16 <- lane15, lane0 <- prev chunk lane31).
      float epm[NCHUNK];
#pragma unroll
      for (int k = 0; k < NCHUNK; ++k) {
        float sh  = dpp_f<0x111>(cm[j][k]);
        float l15 = readlane_f(cm[j][k], 15);
        sh = (lane == 16) ? l15 : sh;
        float pl = (k > 0) ? readlane_f(cm[j][k - 1], 31) : NEGV;
        sh = (lane == 0) ? pl : sh;
        epm[k] = sh;
      }

      // update colmax with inclusive row cummax of v (carry across chunks
      // via readlane(31) -> scalar operand of v_max_f32)
      float carry = NEGV;
#pragma unroll
      for (int k = 0; k < NCHUNK; ++k) {
        float s = wave_incl_cummax(v[k], lane);
        s = fmaxf(s, carry);
        carry = readlane_f(s, 31);
        cm[j][k] = fmaxf(cm[j][k], s);
      }

      // acc[par[j]] += epm  (par is wave-uniform; static dispatch keeps regs)
      const int p = par[j];
#pragma unroll
      for (int q = 0; q < T_NODES - 1; ++q) {
        if (p == q) {
#pragma unroll
          for (int k = 0; k < NCHUNK; ++k) acc[q][k] += epm[k];
        }
      }
    }

    // root node 0: fold row max into running output max
#pragma unroll
    for (int k = 0; k < NCHUNK; ++k)
      outm = fmaxf(outm, (xv[k] + a[0]) + acc[0][k]);
  }

  wait_async_0();

  // full-wave max: inclusive scan puts the total in lane 31
  outm = wave_incl_cummax(outm, lane);
  float total = readlane_f(outm, 31);

  if (lane == 0)
    out[(b * T_TREES + t) * T_C + c] = expm1f(total);
}

// ---------------------------------------------------------------------------
// Host: replicate np.random.default_rng(0) -> make_forest() parents
// (SeedSequence + PCG64 XSL-RR 128 + buffered 32-bit Lemire bounded ints)
// ---------------------------------------------------------------------------
static inline uint64_t mulhi64(uint64_t a, uint64_t b) {
  uint64_t al = (uint32_t)a, ah = a >> 32;
  uint64_t bl = (uint32_t)b, bh = b >> 32;
  uint64_t p0 = al * bl, p1 = al * bh, p2 = ah * bl, p3 = ah * bh;
  uint64_t mid = p1 + (p0 >> 32);
  uint64_t mid2 = p2 + (uint32_t)mid;
  return p3 + (mid >> 32) + (mid2 >> 32);
}

struct NpRng {
  uint64_t shi, slo, ihi, ilo;  // 128-bit state / inc
  uint32_t buf;
  bool has;
};

static void pcg_step(NpRng& r) {
  const uint64_t mh = 2549297995355413924ULL, ml = 4865540595714422341ULL;
  uint64_t lo = r.slo * ml;
  uint64_t hi = mulhi64(r.slo, ml) + r.shi * ml + r.slo * mh;
  uint64_t nlo = lo + r.ilo;
  uint64_t nhi = hi + r.ihi + (nlo < lo ? 1u : 0u);
  r.slo = nlo; r.shi = nhi;
}
static uint64_t pcg_next64(NpRng& r) {
  pcg_step(r);
  uint64_t rot = r.shi >> 58;               // state >> 122
  uint64_t x = r.shi ^ r.slo;
  return rot ? ((x >> rot) | (x << (64 - rot))) : x;
}
static uint32_t np_next32(NpRng& r) {
  if (r.has) { r.has = false; return r.buf; }
  uint64_t n = pcg_next64(r);
  r.has = true; r.buf = (uint32_t)(n >> 32);
  return (uint32_t)n;
}
static uint32_t lemire32(NpRng& r, uint32_t rng /*inclusive max*/) {
  uint32_t rng_excl = rng + 1u;
  uint64_t m = (uint64_t)np_next32(r) * rng_excl;
  uint32_t leftover = (uint32_t)m;
  if (leftover < rng_excl) {
    uint32_t threshold = (uint32_t)((0xFFFFFFFFu - rng) % rng_excl);
    while (leftover < threshold) {
      m = (uint64_t)np_next32(r) * rng_excl;
      leftover = (uint32_t)m;
    }
  }
  return (uint32_t)(m >> 32);
}

static uint32_t ss_hash(uint32_t v, uint32_t& hc) {
  v ^= hc;
  hc *= 0x931e8875u;      // MULT_A
  v *= hc;
  v ^= v >> 16;
  return v;
}
static uint32_t ss_mix(uint32_t x, uint32_t y) {
  uint32_t r = x * 0xca01f9ddu;   // MIX_MULT_L
  r -= y * 0x4973f715u;           // MIX_MULT_R
  r ^= r >> 16;
  return r;
}

static ForestParams make_forest_host() {
  // SeedSequence(0): entropy = [0], pool of 4 uint32
  uint32_t pool[4];
  uint32_t hc = 0x43b0d7e5u;      // INIT_A
  for (int i = 0; i < 4; ++i) pool[i] = ss_hash(0u, hc);
  for (int s = 0; s < 4; ++s)
    for (int d = 0; d < 4; ++d)
      if (s != d) pool[d] = ss_mix(pool[d], ss_hash(pool[s], hc));

  // generate_state(4, uint64) -> 8 uint32 words
  uint32_t hb = 0x8b51f9ddu;      // INIT_B
  uint32_t st[8];
  for (int i = 0; i < 8; ++i) {
    uint32_t dv = pool[i & 3];
    dv ^= hb;
    hb *= 0x58f38dedu;            // MULT_B
    dv *= hb;
    dv ^= dv >> 16;
    st[i] = dv;
  }
  uint64_t w0 = st[0] | ((uint64_t)st[1] << 32);  // initstate hi
  uint64_t w1 = st[2] | ((uint64_t)st[3] << 32);  // initstate lo
  uint64_t w2 = st[4] | ((uint64_t)st[5] << 32);  // initseq  hi
  uint64_t w3 = st[6] | ((uint64_t)st[7] << 32);  // initseq  lo

  NpRng r;
  r.ihi = (w2 << 1) | (w3 >> 63);
  r.ilo = (w3 << 1) | 1ull;
  r.shi = 0; r.slo = 0;
  pcg_step(r);                                   // state = inc
  uint64_t nlo = r.slo + w1;
  r.shi += w0 + (nlo < r.slo ? 1u : 0u);         // state += initstate
  r.slo = nlo;
  pcg_step(r);
  r.has = false; r.buf = 0;

  ForestParams fp;
  for (int t = 0; t < T_TREES; ++t) {
    fp.par[t][0] = -1;
    for (int i = 1; i < T_NODES; ++i) {
      uint32_t rng = (uint32_t)(i - 1);          // integers(0, i): range i-1
      fp.par[t][i] = (rng == 0) ? 0 : (int)lemire32(r, rng);
    }
  }
  return fp;
}

// ---------------------------------------------------------------------------
extern "C" void kernel_launch(void* const* d_in, const int* in_sizes, int n_in,
                              void* d_out, int out_size, void* d_ws, size_t ws_size,
                              hipStream_t stream) {
  (void)in_sizes; (void)n_in; (void)d_ws; (void)ws_size; (void)out_size;
  const float* x      = (const float*)d_in[0];   // (8,16,224,224) f32
  const float* alphas = (const float*)d_in[1];   // (8,5,16) f32
  float* out          = (float*)d_out;           // (8,8,16) f32

  ForestParams fp = make_forest_host();          // deterministic, host-only

  dim3 grid(T_B * T_C);   // 128 planes
  dim3 block(256);        // 8 waves = 8 trees per plane
  fis_forest_kernel<<<grid, block, 0, stream>>>(x, alphas, out, fp);
}